// PairwiseLMFMap_34832184770641
// MI455X (gfx1250) — compile-verified
//
#include <hip/hip_runtime.h>
#include <hip/hip_bf16.h>

// ---- problem constants (match reference) ----
#define BATCH 2
#define LSEQ  512
#define DXY   256      // DX == DY == 256
#define RANK  16
#define DOUT  128
#define EPSV  1e-5f

typedef __attribute__((ext_vector_type(2))) float v2f;
typedef __attribute__((ext_vector_type(4))) float v4f;
typedef __attribute__((ext_vector_type(8))) float v8f;

// D = A(16x4 f32) * B(4x16 f32) + C(16x16 f32), full-precision WMMA on CDNA5.
static __device__ inline v8f wmma4(v2f a, v2f b, v8f c) {
    return __builtin_amdgcn_wmma_f32_16x16x4_f32(
        /*neg_a=*/false, a, /*neg_b=*/false, b,
        /*c_mod=*/(short)0, c, /*reuse_a=*/false, /*reuse_b=*/false);
}

// ---------------------------------------------------------------------------
// Kernel 1: row LayerNorm over 256 features. grid = rows, block = 256.
// ---------------------------------------------------------------------------
__global__ __launch_bounds__(256)
void ln_kernel(const float* __restrict__ x, const float* __restrict__ w,
               const float* __restrict__ b, float* __restrict__ out) {
    __shared__ float s1[256];
    __shared__ float s2[256];
    const int row = blockIdx.x;
    const int t   = threadIdx.x;
    const float v = x[row * DXY + t];
    s1[t] = v;
    s2[t] = v * v;
    __syncthreads();
    for (int off = 128; off > 0; off >>= 1) {
        if (t < off) { s1[t] += s1[t + off]; s2[t] += s2[t + off]; }
        __syncthreads();
    }
    const float mu  = s1[0] * (1.0f / DXY);
    const float var = s2[0] * (1.0f / DXY) - mu * mu;
    const float inv = rsqrtf(var + EPSV);
    out[row * DXY + t] = (v - mu) * inv * w[t] + b[t];
}

// ---------------------------------------------------------------------------
// Kernel 2: projection GEMM  out[i, c] = xn[i,:] . W[c,:] + bias[c]
//   xn: [1024 x 256], W: [2048 x 256] (torch [out,in]), c = r*128 + o.
//   Stored transposed as outT[b][o][i][r] (b = i/512) so kernel 3 gets
//   contiguous operand loads.
//   One wave -> one 16x16 tile, K-loop = 256/4 = 64 f32 WMMAs.
//   block = 128 (4 waves, 4 column tiles), grid = (2048/64, 1024/16).
// ---------------------------------------------------------------------------
__global__ __launch_bounds__(128)
void proj_kernel(const float* __restrict__ xn, const float* __restrict__ W,
                 const float* __restrict__ bias, float* __restrict__ outT) {
    const int wave = threadIdx.x >> 5;
    const int lane = threadIdx.x & 31;
    const int m    = lane & 15;     // A-row / B-col held by this lane
    const int hi   = lane >> 4;     // 0: K0,K1   1: K2,K3

    const int c0 = (blockIdx.x * 4 + wave) * 16;   // output column tile
    const int i0 = blockIdx.y * 16;                // global row tile (b*512+i)

    const float* aRow = xn + (i0 + m) * DXY + hi * 2;
    const float* bRow = W  + (c0 + m) * DXY + hi * 2;

    v8f acc = {0.f, 0.f, 0.f, 0.f, 0.f, 0.f, 0.f, 0.f};
#pragma unroll 8
    for (int k = 0; k < DXY / 4; ++k) {
        v2f av = *(const v2f*)(aRow + 4 * k);
        v2f bv = *(const v2f*)(bRow + 4 * k);
        acc = wmma4(av, bv, acc);
    }

    const float bc = bias[c0 + m];
    const int o  = (c0 + m) & (DOUT - 1);   // channel
    const int r  = c0 >> 7;                 // rank index (tile never crosses)
    const int bb = i0 >> 9;                 // batch
    const int il = i0 & (LSEQ - 1);
#pragma unroll
    for (int v = 0; v < 8; ++v) {
        const int irow = il + v + hi * 8;   // C/D layout: M = v + 8*hi
        outT[((size_t)(bb * DOUT + o) * LSEQ + irow) * RANK + r] = acc[v] + bc;
    }
}

// ---------------------------------------------------------------------------
// Kernel 3: pairwise rank contraction + channel LayerNorm.
//   out[b,o,i,j] = LN_o( sum_r aT[b,o,i,r] * bT[b,o,j,r] )
//   grid = (32 j-tiles, 32 i-tiles, 2 batches), block = 256 (8 waves).
//   Wave w computes channels o = 16w..16w+15 (4 f32 WMMAs each),
//   tiles staged in 129-stride LDS (bank-conflict-free column reads).
//   Stats pass: one thread per (i,j) position -> mu/inv into LDS.
//   Store pass: float4 non-temporal stores along j (output is 268 MB,
//   write-once -> bypass L2 so WMMA operand re-reads stay resident).
// ---------------------------------------------------------------------------
__global__ __launch_bounds__(256)
void pair_kernel(const float* __restrict__ aT, const float* __restrict__ bT,
                 const float* __restrict__ ow, const float* __restrict__ ob,
                 float* __restrict__ out) {
    __shared__ float lds[256 * 129];   // 132 KB of the WGP's 320 KB LDS
    __shared__ float smu[256];
    __shared__ float sinv[256];

    const int t    = threadIdx.x;
    const int wave = t >> 5;
    const int lane = t & 31;
    const int m    = lane & 15;
    const int hi   = lane >> 4;

    const int j0 = blockIdx.x * 16;
    const int i0 = blockIdx.y * 16;
    const int bb = blockIdx.z;

    for (int c = 0; c < 16; ++c) {
        const int o = wave * 16 + c;
        const float* aBase =
            aT + ((size_t)(bb * DOUT + o) * LSEQ + i0 + m) * RANK + hi * 2;
        const float* bBase =
            bT + ((size_t)(bb * DOUT + o) * LSEQ + j0 + m) * RANK + hi * 2;

        v8f acc = {0.f, 0.f, 0.f, 0.f, 0.f, 0.f, 0.f, 0.f};
#pragma unroll
        for (int k = 0; k < RANK / 4; ++k) {
            v2f av = *(const v2f*)(aBase + 4 * k);
            v2f bv = *(const v2f*)(bBase + 4 * k);
            acc = wmma4(av, bv, acc);
        }
#pragma unroll
        for (int v = 0; v < 8; ++v) {
            const int pos = (v + hi * 8) * 16 + m;  // i_local*16 + j_local
            lds[pos * 129 + o] = acc[v];
        }
    }
    __syncthreads();

    // ---- stats pass: one thread per spatial position ----
    {
        const float* row = lds + t * 129;
        float s = 0.f, s2 = 0.f;
#pragma unroll 8
        for (int o = 0; o < DOUT; ++o) {
            const float v = row[o];
            s += v; s2 += v * v;
        }
        const float mu  = s * (1.0f / DOUT);
        const float var = s2 * (1.0f / DOUT) - mu * mu;
        smu[t]  = mu;
        sinv[t] = rsqrtf(var + EPSV);
    }
    __syncthreads();

    // ---- store pass: float4 NT stores along j ----
    // 128 o x 16 i x 4 j-quads = 8192 float4 per tile, 32 per thread.
    for (int idx = t; idx < DOUT * 16 * 4; idx += 256) {
        const int o   = idx >> 6;         // 0..127
        const int rem = idx & 63;
        const int il  = rem >> 2;         // 0..15
        const int jq  = rem & 3;          // 0..3 (quad of j)
        const int pos = il * 16 + jq * 4;

        const float wo = ow[o];
        const float bo = ob[o];
        v4f r;
        r.x = (lds[(pos + 0) * 129 + o] - smu[pos + 0]) * sinv[pos + 0] * wo + bo;
        r.y = (lds[(pos + 1) * 129 + o] - smu[pos + 1]) * sinv[pos + 1] * wo + bo;
        r.z = (lds[(pos + 2) * 129 + o] - smu[pos + 2]) * sinv[pos + 2] * wo + bo;
        r.w = (lds[(pos + 3) * 129 + o] - smu[pos + 3]) * sinv[pos + 3] * wo + bo;

        float* dst = out + ((size_t)(bb * DOUT + o) * LSEQ + (i0 + il)) * LSEQ
                         + (j0 + jq * 4);
        __builtin_nontemporal_store(r, (v4f*)dst);
    }
}

// ---------------------------------------------------------------------------
// Host entry
// ---------------------------------------------------------------------------
extern "C" void kernel_launch(void* const* d_in, const int* in_sizes, int n_in,
                              void* d_out, int out_size, void* d_ws, size_t ws_size,
                              hipStream_t stream) {
    const float* x   = (const float*)d_in[0];
    const float* y   = (const float*)d_in[1];
    const float* xnw = (const float*)d_in[2];
    const float* xnb = (const float*)d_in[3];
    const float* ynw = (const float*)d_in[4];
    const float* ynb = (const float*)d_in[5];
    const float* Wx  = (const float*)d_in[6];
    const float* bx  = (const float*)d_in[7];
    const float* Wy  = (const float*)d_in[8];
    const float* by  = (const float*)d_in[9];
    const float* ow  = (const float*)d_in[10];
    const float* ob  = (const float*)d_in[11];
    float* out = (float*)d_out;

    // workspace layout (floats):
    //   xn : 1024*256          = 262144
    //   yn : 262144
    //   aT : 2*128*512*16      = 2097152   [b][o][i][r]
    //   bT : 2097152
    float* ws = (float*)d_ws;
    float* xn = ws;
    float* yn = ws + 262144;
    float* aT = ws + 524288;
    float* bT = ws + 2621440;

    const int rows = BATCH * LSEQ;  // 1024

    ln_kernel<<<rows, 256, 0, stream>>>(x, xnw, xnb, xn);
    ln_kernel<<<rows, 256, 0, stream>>>(y, ynw, ynb, yn);

    dim3 pgrid(RANK * DOUT / 64, rows / 16);  // (32, 64)
    proj_kernel<<<pgrid, 128, 0, stream>>>(xn, Wx, bx, aT);
    proj_kernel<<<pgrid, 128, 0, stream>>>(yn, Wy, by, bT);

    dim3 fgrid(LSEQ / 16, LSEQ / 16, BATCH);  // (32, 32, 2)
    pair_kernel<<<fgrid, 256, 0, stream>>>(aT, bT, ow, ob, out);
}